// WindowAttention_68289980007064
// MI455X (gfx1250) — compile-verified
//
#include <hip/hip_runtime.h>
#include <math.h>

typedef __attribute__((ext_vector_type(2))) float v2f;
typedef __attribute__((ext_vector_type(8))) float v8f;

#define WSZ   4
#define DISP  2
#define HEADS 3
#define HD    32
#define DIM   96            // input/output channels
#define INNER 96            // HEADS*HD
#define N3    288           // 3*INNER
#define IMG   112
#define NWH   28            // IMG/WSZ
#define PIX   16            // WSZ*WSZ

// padded LDS row strides (odd -> conflict-free column reads across 16 lanes)
#define XW_S   (DIM + 1)    // 97
#define QKV_S  (N3 + 1)     // 289
#define ATT_S  (PIX + 1)    // 17
#define OY_S   (INNER + 1)  // 97

static __device__ __forceinline__ v8f wmma4f32(v2f a, v2f b, v8f c) {
    // V_WMMA_F32_16X16X4_F32 : D = A(16x4) * B(4x16) + C, fp32 throughout
    return __builtin_amdgcn_wmma_f32_16x16x4_f32(
        /*neg_a=*/false, a, /*neg_b=*/false, b,
        /*c_mod=*/(short)0, c, /*reuse_a=*/false, /*reuse_b=*/false);
}

__global__ __launch_bounds__(128)
void swin_window_attn_kernel(const float* __restrict__ x,
                             const float* __restrict__ Wqkv,
                             const float* __restrict__ pe,
                             const float* __restrict__ Wout,
                             const float* __restrict__ bout,
                             float* __restrict__ out)
{
    __shared__ float xw  [PIX * XW_S];          // shifted input window 16x96
    __shared__ float qkv [PIX * QKV_S];         // 16x288 (q|k|v)
    __shared__ float attn[HEADS * PIX * ATT_S]; // per-head 16x16 attention
    __shared__ float oy  [PIX * OY_S];          // attention output 16x96
    __shared__ float pes [49];                  // 7x7 relative position table

    const int tid  = threadIdx.x;
    const int lane = tid & 31;
    const int wave = tid >> 5;
    const int col  = lane & 15;      // N (or M for A-frags)
    const int half = lane >> 4;
    const int koff = half * 2;       // K offset of this half-wave's fragment

    const int win = blockIdx.x % (NWH * NWH);
    const int b   = blockIdx.x / (NWH * NWH);
    const int wr  = win / NWH;
    const int wc  = win % NWH;

    // ---------------- stage rolled(-2,-2) window + pos table ----------------
    for (int idx = tid; idx < PIX * DIM; idx += 128) {
        int p  = idx / DIM;
        int ch = idx - p * DIM;
        int r  = wr * WSZ + (p >> 2);
        int c  = wc * WSZ + (p & 3);
        int gr = (r + DISP) % IMG;           // shifted[r] = x[(r+2)%112]
        int gc = (c + DISP) % IMG;
        xw[p * XW_S + ch] =
            x[(((size_t)b * IMG + gr) * IMG + gc) * DIM + ch];
    }
    if (tid < 49) pes[tid] = pe[tid];
    __syncthreads();

    // ---------------- GEMM1: qkv(16x288) = xw(16x96) @ Wqkv(96x288) ---------
    for (int t = wave; t < N3 / 16; t += 4) {
        const int n0 = t * 16;
        v8f acc = {};
        #pragma unroll
        for (int ks = 0; ks < DIM / 4; ++ks) {
            const int k0 = ks * 4 + koff;
            v2f a, bm;
            a.x  = xw[col * XW_S + k0];
            a.y  = xw[col * XW_S + k0 + 1];
            bm.x = Wqkv[(size_t)k0       * N3 + n0 + col];
            bm.y = Wqkv[(size_t)(k0 + 1) * N3 + n0 + col];
            acc = wmma4f32(a, bm, acc);
        }
        #pragma unroll
        for (int r = 0; r < 8; ++r)
            qkv[(r + 8 * half) * QKV_S + n0 + col] = acc[r];
    }
    __syncthreads();

    // ---------------- attention: one wave per head --------------------------
    if (wave < HEADS) {
        const int h = wave;
        const float scale = 0.17677669529663687f;   // 1/sqrt(32)

        // dots = q @ k^T   (16x16, K=32 -> 8 WMMAs)
        v8f d = {};
        #pragma unroll
        for (int ks = 0; ks < HD / 4; ++ks) {
            const int k0 = ks * 4 + koff;
            v2f a, bm;
            a.x  = qkv[col * QKV_S + h * HD + k0];
            a.y  = qkv[col * QKV_S + h * HD + k0 + 1];
            bm.x = qkv[col * QKV_S + INNER + h * HD + k0];
            bm.y = qkv[col * QKV_S + INNER + h * HD + k0 + 1];
            d = wmma4f32(a, bm, d);
        }

        // scale + relative-position bias + shift masks, then row softmax.
        // C layout: element (i = r + 8*half, j = col); row i spans the 16
        // lanes of this half-wave -> reduce with shfl_xor masks 1,2,4,8.
        const bool bot = (wr == NWH - 1);
        const bool rgt = (wc == NWH - 1);
        #pragma unroll
        for (int r = 0; r < 8; ++r) {
            const int i  = r + 8 * half;
            const int j  = col;
            const int xi = i >> 2, yi = i & 3;
            const int xj = j >> 2, yj = j & 3;
            float v = d[r] * scale + pes[(xj - xi + 3) * 7 + (yj - yi + 3)];
            if (bot && ((i >= 8)  != (j >= 8)))  v = -3.0e38f;  // upper/lower mask
            if (rgt && ((yi >= 2) != (yj >= 2))) v = -3.0e38f;  // left/right mask

            float m = v;
            m = fmaxf(m, __shfl_xor(m, 1, 32));
            m = fmaxf(m, __shfl_xor(m, 2, 32));
            m = fmaxf(m, __shfl_xor(m, 4, 32));
            m = fmaxf(m, __shfl_xor(m, 8, 32));
            float e = __expf(v - m);
            float s = e;
            s += __shfl_xor(s, 1, 32);
            s += __shfl_xor(s, 2, 32);
            s += __shfl_xor(s, 4, 32);
            s += __shfl_xor(s, 8, 32);
            attn[(h * PIX + i) * ATT_S + j] = e / s;
        }

        // out_h(16x32) = attn(16x16) @ v(16x32) : 2 N-tiles x 4 K-steps
        #pragma unroll
        for (int nt = 0; nt < 2; ++nt) {
            const int n0 = nt * 16;
            v8f acc = {};
            #pragma unroll
            for (int ks = 0; ks < 4; ++ks) {
                const int k0 = ks * 4 + koff;
                v2f a, bm;
                a.x  = attn[(h * PIX + col) * ATT_S + k0];
                a.y  = attn[(h * PIX + col) * ATT_S + k0 + 1];
                bm.x = qkv[k0       * QKV_S + 2 * INNER + h * HD + n0 + col];
                bm.y = qkv[(k0 + 1) * QKV_S + 2 * INNER + h * HD + n0 + col];
                acc = wmma4f32(a, bm, acc);
            }
            #pragma unroll
            for (int r = 0; r < 8; ++r)
                oy[(r + 8 * half) * OY_S + h * HD + n0 + col] = acc[r];
        }
    }
    __syncthreads();

    // ---------------- GEMM2: y(16x96) = oy(16x96) @ Wout(96x96) + b ---------
    for (int t = wave; t < INNER / 16; t += 4) {
        const int n0 = t * 16;
        v8f acc = {};
        #pragma unroll
        for (int ks = 0; ks < DIM / 4; ++ks) {
            const int k0 = ks * 4 + koff;
            v2f a, bm;
            a.x  = oy[col * OY_S + k0];
            a.y  = oy[col * OY_S + k0 + 1];
            bm.x = Wout[(size_t)k0       * DIM + n0 + col];
            bm.y = Wout[(size_t)(k0 + 1) * DIM + n0 + col];
            acc = wmma4f32(a, bm, acc);
        }
        const float bb = bout[n0 + col];   // N is fixed per lane
        #pragma unroll
        for (int r = 0; r < 8; ++r) {
            const int i  = r + 8 * half;
            const int rr = wr * WSZ + (i >> 2);
            const int cc = wc * WSZ + (i & 3);
            const int gr = (rr + DISP) % IMG;   // roll back (+2,+2)
            const int gc = (cc + DISP) % IMG;
            out[(((size_t)b * IMG + gr) * IMG + gc) * DIM + n0 + col] = acc[r] + bb;
        }
    }
}

extern "C" void kernel_launch(void* const* d_in, const int* in_sizes, int n_in,
                              void* d_out, int out_size, void* d_ws, size_t ws_size,
                              hipStream_t stream) {
    const float* x    = (const float*)d_in[0];
    const float* Wqkv = (const float*)d_in[1];
    const float* pe   = (const float*)d_in[2];
    const float* Wout = (const float*)d_in[3];
    const float* bout = (const float*)d_in[4];
    float* out = (float*)d_out;

    const int batch = in_sizes[0] / (IMG * IMG * DIM);   // 32
    dim3 grid((unsigned)(batch * NWH * NWH));            // one block per (b, window)
    swin_window_attn_kernel<<<grid, 128, 0, stream>>>(x, Wqkv, pe, Wout, bout, out);
}